// FNOBlock_4501125726908
// MI455X (gfx1250) — compile-verified
//
#include <hip/hip_runtime.h>
#include <math.h>

typedef float v2f __attribute__((ext_vector_type(2)));
typedef float v8f __attribute__((ext_vector_type(8)));

#define TWO_PI 6.28318530717958647692f

// workspace layout (float offsets)
#define WT_OFF  0ull
#define WT_SZ   (24ull*12*64*64*2)        // transposed weights [ky24][kx12][i][o][2]
#define XW_OFF  (WT_OFF + WT_SZ)
#define XW_SZ   (16ull*64*256*12*2)       // row-DFT  [b][c][h][kx][2]
#define XF_OFF  (XW_OFF + XW_SZ)
#define XF_SZ   (24ull*12*16*64*2)        // modes    [ky][kx][b][i][2]
#define OF_OFF  (XF_OFF + XF_SZ)
#define OF_SZ   XF_SZ                     // mixed    [ky][kx][b][o][2]
#define YS_OFF  (OF_OFF + OF_SZ)
#define YS_SZ   (16ull*256*64*12*2)       // inv-col  [b][h][o][kx][2]
#define ST_OFF  (YS_OFF + YS_SZ)          // sum[64], sumsq[64]
#define CO_OFF  (ST_OFF + 128ull)         // a[64], b[64]

// ---------------- kernel 0: zero stats + transpose spectral weights ----------------
__global__ __launch_bounds__(256) void k_prep(const float* __restrict__ w1,
                                              const float* __restrict__ w2,
                                              float* __restrict__ ws) {
  long long idx = (long long)blockIdx.x * 256 + threadIdx.x;
  if (blockIdx.x == 0 && threadIdx.x < 128) ws[ST_OFF + threadIdx.x] = 0.f;
  const long long N = 24ll * 12 * 64 * 64;
  if (idx >= N) return;
  int o = (int)(idx & 63);
  int i = (int)((idx >> 6) & 63);
  long long q = idx >> 12;          // kyi*12 + kx
  int kx  = (int)(q % 12);
  int kyi = (int)(q / 12);
  const float* src = (kyi < 12) ? w1 : w2;
  int y = (kyi < 12) ? kyi : kyi - 12;
  long long s = ((((long long)i * 64 + o) * 12 + y) * 12 + kx) * 2;
  ws[WT_OFF + idx * 2]     = src[s];
  ws[WT_OFF + idx * 2 + 1] = src[s + 1];
}

// ---------------- kernel 1: row DFT (keep kx 0..11), one wave per row ----------------
__global__ __launch_bounds__(256) void k_rowdft(const float* __restrict__ x,
                                                float* __restrict__ ws) {
  __shared__ float2 tw[256];
  int tid = threadIdx.x;
  { float s, c; sincosf(-TWO_PI * (float)tid / 256.f, &s, &c); tw[tid] = make_float2(c, s); }
  __syncthreads();
  int wave = tid >> 5, lane = tid & 31;
  long long row = (long long)blockIdx.x * 8 + wave;   // (b*64+c)*256 + h
  const float* xr = x + row * 256;
  float xv[8];
#pragma unroll
  for (int j = 0; j < 8; j++) xv[j] = xr[lane + 32 * j];
  float re[12] = {0.f}, im[12] = {0.f};
#pragma unroll
  for (int j = 0; j < 8; j++) {
    int w = lane + 32 * j;
    float v = xv[j];
#pragma unroll
    for (int kx = 0; kx < 12; kx++) {
      float2 t = tw[(kx * w) & 255];
      re[kx] += v * t.x; im[kx] += v * t.y;
    }
  }
#pragma unroll
  for (int kx = 0; kx < 12; kx++) {
#pragma unroll
    for (int m = 16; m > 0; m >>= 1) {
      re[kx] += __shfl_xor(re[kx], m, 32);
      im[kx] += __shfl_xor(im[kx], m, 32);
    }
  }
  if (lane == 0) {
    float* dst = ws + XW_OFF + row * 24;
#pragma unroll
    for (int kx = 0; kx < 12; kx++) { dst[kx * 2] = re[kx]; dst[kx * 2 + 1] = im[kx]; }
  }
}

// ---------------- kernel 2: column DFT at ky in {0..11, 244..255} ----------------
__global__ __launch_bounds__(256) void k_coldft(float* __restrict__ ws) {
  __shared__ float2 tw[256];
  __shared__ float sxw[6144];        // [h=256][kx=12][2]
  int tid = threadIdx.x;
  { float s, c; sincosf(-TWO_PI * (float)tid / 256.f, &s, &c); tw[tid] = make_float2(c, s); }
  int bc_ = blockIdx.x;              // b*64 + cin
  const float* src = ws + XW_OFF + (long long)bc_ * 6144;
  for (int e = tid; e < 6144; e += 256) sxw[e] = src[e];
  __syncthreads();
  int b = bc_ >> 6, ci = bc_ & 63;
  for (int o = tid; o < 288; o += 256) {
    int kyi = o / 12, kx = o % 12;
    int ky = (kyi < 12) ? kyi : kyi + 232;    // 244..255
    float ar = 0.f, ai = 0.f;
    for (int h = 0; h < 256; h++) {
      float2 t = tw[(ky * h) & 255];
      float xr = sxw[(h * 12 + kx) * 2], xi = sxw[(h * 12 + kx) * 2 + 1];
      ar += xr * t.x - xi * t.y;
      ai += xr * t.y + xi * t.x;
    }
    float* dst = ws + XF_OFF + (((long long)o * 16 + b) * 64 + ci) * 2;
    dst[0] = ar; dst[1] = ai;
  }
}

// ---------------- kernel 3: complex mode mixing, WMMA f32 16x16x4 ----------------
// per (ky,kx): D[b(16), o(64)] = sum_i A[b,i] * W[i,o]  (complex)
__global__ __launch_bounds__(128) void k_modemix(float* __restrict__ ws) {
  __shared__ float Are[16 * 65], Aim[16 * 65];
  __shared__ float Bre[64 * 72], Bim[64 * 72];
  int tid = threadIdx.x;
  int blk = blockIdx.x;              // kyi*12 + kx
  const float* fa = ws + XF_OFF + (long long)blk * 2048;
  for (int e = tid; e < 1024; e += 128) {
    int bb = e >> 6, ii = e & 63;
    Are[bb * 65 + ii] = fa[e * 2]; Aim[bb * 65 + ii] = fa[e * 2 + 1];
  }
  const float* fb = ws + WT_OFF + (long long)blk * 8192;
  for (int e = tid; e < 4096; e += 128) {
    int kk = e >> 6, nn = e & 63;
    Bre[kk * 72 + nn] = fb[e * 2]; Bim[kk * 72 + nn] = fb[e * 2 + 1];
  }
  __syncthreads();
  int wave = tid >> 5, lane = tid & 31, half = lane >> 4, ln = lane & 15;
  int nb = wave * 16;                // o-tile
  v8f prr = {}, pii = {}, pri = {}, pir = {};
#pragma unroll
  for (int kk = 0; kk < 16; kk++) {
    int kA = kk * 4 + half * 2;
    v2f ar, ai, br, bi;
    ar[0] = Are[ln * 65 + kA];      ar[1] = Are[ln * 65 + kA + 1];
    ai[0] = Aim[ln * 65 + kA];      ai[1] = Aim[ln * 65 + kA + 1];
    br[0] = Bre[kA * 72 + nb + ln]; br[1] = Bre[(kA + 1) * 72 + nb + ln];
    bi[0] = Bim[kA * 72 + nb + ln]; bi[1] = Bim[(kA + 1) * 72 + nb + ln];
    prr = __builtin_amdgcn_wmma_f32_16x16x4_f32(false, ar, false, br, (short)0, prr, false, false);
    pii = __builtin_amdgcn_wmma_f32_16x16x4_f32(false, ai, false, bi, (short)0, pii, false, false);
    pri = __builtin_amdgcn_wmma_f32_16x16x4_f32(false, ar, false, bi, (short)0, pri, false, false);
    pir = __builtin_amdgcn_wmma_f32_16x16x4_f32(false, ai, false, br, (short)0, pir, false, false);
  }
  float* dst = ws + OF_OFF + (long long)blk * 2048;
#pragma unroll
  for (int r = 0; r < 8; r++) {
    int bidx = r + half * 8;
    int o = nb + ln;
    float* p = dst + (bidx * 64 + o) * 2;
    p[0] = prr[r] - pii[r];
    p[1] = pri[r] + pir[r];
  }
}

// ---------------- kernel 4: inverse column transform (24 ky -> 256 h) ----------------
__global__ __launch_bounds__(256) void k_invcol(float* __restrict__ ws) {
  __shared__ float2 tw[256];
  __shared__ float2 sof[24 * 12 * 8];
  int tid = threadIdx.x;
  { float s, c; sincosf(TWO_PI * (float)tid / 256.f, &s, &c); tw[tid] = make_float2(c, s); }
  int b = blockIdx.x >> 3, o0 = (blockIdx.x & 7) * 8;
  const float* src = ws + OF_OFF;
  for (int e = tid; e < 2304; e += 256) {
    int kyi = e / 96, rr = e % 96, kx = rr >> 3, oo = rr & 7;
    const float* p = src + (((long long)(kyi * 12 + kx) * 16 + b) * 64 + o0 + oo) * 2;
    sof[e] = make_float2(p[0], p[1]);
  }
  __syncthreads();
  for (int e = tid; e < 24576; e += 256) {
    int h = e / 96, rr = e % 96, oo = rr / 12, kx = rr % 12;
    float ar = 0.f, ai = 0.f;
#pragma unroll
    for (int kyi = 0; kyi < 24; kyi++) {
      int ky = (kyi < 12) ? kyi : kyi + 232;
      float2 t = tw[(ky * h) & 255];
      float2 F = sof[(kyi * 12 + kx) * 8 + oo];
      ar += F.x * t.x - F.y * t.y;
      ai += F.x * t.y + F.y * t.x;
    }
    float sc = ((kx == 0) ? 1.f : 2.f) / 65536.f;   // irfft2 norm + hermitian doubling
    float* dst = ws + YS_OFF + ((((long long)b * 256 + h) * 64 + (o0 + oo)) * 12 + kx) * 2;
    dst[0] = ar * sc; dst[1] = ai * sc;
  }
}

// ---------------- kernel 5: fused inverse row-DFT + 1x1 conv (WMMA) + bias + stats ----
__global__ __launch_bounds__(256) void k_fused(const float* __restrict__ x,
                                               const float* __restrict__ wc,
                                               const float* __restrict__ bc,
                                               float* __restrict__ out,
                                               float* __restrict__ ws) {
  __shared__ float2 tw[256];
  __shared__ float swc[64 * 65];     // wc[o][i]
  __shared__ float sx[64 * 136];     // x[i][p], p = 128 w-points
  __shared__ float sys[1536];        // Ys[o][kx][2]
  __shared__ float sbias[64];
  __shared__ float lsum[64], lsq[64];
  int tid = threadIdx.x;
  int wck = blockIdx.x, h = blockIdx.y, b = blockIdx.z;
  int w0 = wck * 128;
  { float s, c; sincosf(TWO_PI * (float)tid / 256.f, &s, &c); tw[tid] = make_float2(c, s); }
  for (int e = tid; e < 4096; e += 256) swc[(e >> 6) * 65 + (e & 63)] = wc[e];
  const float* xb = x + (((long long)b * 64) * 256 + h) * 256 + w0;
  for (int e = tid; e < 8192; e += 256) {
    int i = e >> 7, p = e & 127;
    sx[i * 136 + p] = xb[(long long)i * 65536 + p];
  }
  const float* ysb = ws + YS_OFF + (((long long)b * 256 + h) * 64) * 24;
  for (int e = tid; e < 1536; e += 256) sys[e] = ysb[e];
  if (tid < 64) { sbias[tid] = bc[tid]; lsum[tid] = 0.f; lsq[tid] = 0.f; }
  __syncthreads();
  int wave = tid >> 5, lane = tid & 31, half = lane >> 4, ln = lane & 15;
  for (int t = 0; t < 4; t++) {
    int tile = wave * 4 + t;
    int otile = tile >> 3, wtile = tile & 7;
    v8f acc = {};
#pragma unroll
    for (int kk = 0; kk < 16; kk++) {
      int kA = kk * 4 + half * 2;
      v2f a, bfr;
      a[0]   = swc[(otile * 16 + ln) * 65 + kA];
      a[1]   = swc[(otile * 16 + ln) * 65 + kA + 1];
      bfr[0] = sx[kA * 136 + wtile * 16 + ln];
      bfr[1] = sx[(kA + 1) * 136 + wtile * 16 + ln];
      acc = __builtin_amdgcn_wmma_f32_16x16x4_f32(false, a, false, bfr, (short)0, acc, false, false);
    }
    int w = w0 + wtile * 16 + ln;
    float cwv[12], swv[12];
#pragma unroll
    for (int kx = 0; kx < 12; kx++) { float2 tt = tw[(kx * w) & 255]; cwv[kx] = tt.x; swv[kx] = tt.y; }
#pragma unroll
    for (int r = 0; r < 8; r++) {
      int o = otile * 16 + r + half * 8;
      float val = acc[r] + sbias[o];
#pragma unroll
      for (int kx = 0; kx < 12; kx++)
        val += sys[(o * 12 + kx) * 2] * cwv[kx] - sys[(o * 12 + kx) * 2 + 1] * swv[kx];
      out[(((long long)b * 64 + o) * 256 + h) * 256 + w] = val;
      float s1 = val, s2 = val * val;
#pragma unroll
      for (int m = 8; m > 0; m >>= 1) { s1 += __shfl_xor(s1, m, 16); s2 += __shfl_xor(s2, m, 16); }
      if (ln == 0) { atomicAdd(&lsum[o], s1); atomicAdd(&lsq[o], s2); }
    }
  }
  __syncthreads();
  if (tid < 64) {
    atomicAdd(&ws[ST_OFF + tid], lsum[tid]);
    atomicAdd(&ws[ST_OFF + 64 + tid], lsq[tid]);
  }
}

// ---------------- kernel 6: BN coefficients ----------------
__global__ void k_coef(const float* __restrict__ gamma, const float* __restrict__ beta,
                       float* __restrict__ ws) {
  int c = threadIdx.x;
  if (c >= 64) return;
  const float n = 4194304.f;   // 4*B*H*W
  float mean = ws[ST_OFF + c] / n;
  float var  = ws[ST_OFF + 64 + c] / n - mean * mean;
  float a = gamma[c] * rsqrtf(var + 1e-5f);
  ws[CO_OFF + c]      = a;
  ws[CO_OFF + 64 + c] = beta[c] - mean * a;
}

// ---------------- kernel 7: in-place BN + exact GELU ----------------
__global__ __launch_bounds__(256) void k_bngelu(float* __restrict__ out,
                                                const float* __restrict__ ws) {
  long long i4 = ((long long)blockIdx.x * 256 + threadIdx.x) * 4;
  int c = (int)((i4 >> 16) & 63);
  float a = ws[CO_OFF + c], bb = ws[CO_OFF + 64 + c];
  float4 v = *(const float4*)(out + i4);
  float r[4] = {v.x, v.y, v.z, v.w};
#pragma unroll
  for (int j = 0; j < 4; j++) {
    float y = a * r[j] + bb;
    r[j] = 0.5f * y * (1.f + erff(y * 0.70710678118f));
  }
  *(float4*)(out + i4) = make_float4(r[0], r[1], r[2], r[3]);
}

extern "C" void kernel_launch(void* const* d_in, const int* in_sizes, int n_in,
                              void* d_out, int out_size, void* d_ws, size_t ws_size,
                              hipStream_t stream) {
  (void)in_sizes; (void)n_in; (void)out_size; (void)ws_size;
  const float* x     = (const float*)d_in[0];
  const float* w1    = (const float*)d_in[1];
  const float* w2    = (const float*)d_in[2];
  const float* wc    = (const float*)d_in[3];
  const float* bc    = (const float*)d_in[4];
  const float* gamma = (const float*)d_in[5];
  const float* beta  = (const float*)d_in[6];
  float* out = (float*)d_out;
  float* ws  = (float*)d_ws;

  k_prep   <<<9216, 256, 0, stream>>>(w1, w2, ws);
  k_rowdft <<<32768, 256, 0, stream>>>(x, ws);
  k_coldft <<<1024, 256, 0, stream>>>(ws);
  k_modemix<<<288, 128, 0, stream>>>(ws);
  k_invcol <<<128, 256, 0, stream>>>(ws);
  k_fused  <<<dim3(2, 256, 16), 256, 0, stream>>>(x, wc, bc, out, ws);
  k_coef   <<<1, 64, 0, stream>>>(gamma, beta, ws);
  k_bngelu <<<65536, 256, 0, stream>>>(out, ws);
}